// VectorQuantizerEMA_10763188044455
// MI455X (gfx1250) — compile-verified
//
#include <hip/hip_runtime.h>
#include <float.h>
#include <math.h>

typedef __attribute__((ext_vector_type(2))) float v2f;
typedef __attribute__((ext_vector_type(8))) float v8f;
typedef __attribute__((ext_vector_type(4))) unsigned int u32x4;
typedef __attribute__((ext_vector_type(8))) int i32x8;
typedef __attribute__((ext_vector_type(4))) int i32x4;

#define N_ROWS 16384
#define K_CODES 4096
#define DIM 512
#define MT 128          // rows per block in distance kernel
#define NT 128          // columns per accumulation pass
#define DC 32           // D-chunk staged in LDS
#define ST 36           // LDS row stride (floats): 32 data + 4 pad (TDM pad fields)
#define NSPLIT 4        // column splits across grid.y
#define KSPLIT (K_CODES / NSPLIT)
#define NCHUNK (DIM / DC)

// workspace layout (float units)
#define WS_NORMS 0
#define WS_COUNTS 4096
#define WS_LOSS   8192
#define WS_CANDV  8704
#define WS_CANDI  (WS_CANDV + NSPLIT * N_ROWS)
#define WS_IDX    (WS_CANDI + NSPLIT * N_ROWS)

// output layout (floats): z_q_st | indices(as float) | vq_loss | perplexity
#define OUT_IDX  (N_ROWS * DIM)
#define OUT_SCAL (OUT_IDX + N_ROWS)

__global__ void vq_init(unsigned* __restrict__ counts, float* __restrict__ loss) {
  int i = blockIdx.x * blockDim.x + threadIdx.x;
  if (i < K_CODES) counts[i] = 0u;
  if (i == 0) *loss = 0.0f;
}

// one wave per codebook row: ||e_k||^2
__global__ void vq_norms(const float* __restrict__ embed, float* __restrict__ norms) {
  const int w = threadIdx.x >> 5, lane = threadIdx.x & 31;
  const int row = blockIdx.x * 8 + w;
  const float* e = embed + (size_t)row * DIM;
  float s = 0.0f;
#pragma unroll
  for (int i = 0; i < 4; ++i) {
    float4 v = *(const float4*)(e + i * 128 + lane * 4);
    s += v.x * v.x + v.y * v.y + v.z * v.z + v.w * v.w;
  }
#pragma unroll
  for (int m = 16; m >= 1; m >>= 1) s += __shfl_xor(s, m, 32);
  if (lane == 0) norms[row] = s;
}

// Issue one TDM 2D tile load: 128 rows x 32 floats, row stride 512 floats in
// memory, deposited into LDS with a 4-DWORD pad after every 32 DWORDs so the
// LDS row stride is ST=36 floats (matches WMMA fragment indexing).
__device__ __forceinline__ void tdm_load_tile(const float* gptr, unsigned lds_byte,
                                              unsigned tensor_rows) {
  unsigned long long ga = (unsigned long long)gptr;
  u32x4 g0;
  g0.x = 1u;                                    // count=1, is_restore=0, gather=0
  g0.y = lds_byte;                              // D#.lds_addr (bytes)
  g0.z = (unsigned)ga;                          // global_addr[31:0]
  g0.w = (unsigned)((ga >> 32) & 0x01FFFFFFull) // global_addr[56:32]
       | (2u << 30);                            // type = 2 ("image")
  i32x8 g1;
  g1[0] = (int)((2u << 16)                      // data_size: 4 bytes
              | (1u << 20)                      // pad_enable
              | (4u << 22)                      // pad_interval code 4 = 32 DWORDs
              | (3u << 25));                    // pad_amount  code 3 = 4 DWORDs
  g1[1] = (int)(((unsigned)DIM & 0xFFFFu) << 16);              // tensor_dim0 lo16
  g1[2] = (int)(((unsigned)DIM >> 16)
              | ((tensor_rows & 0xFFFFu) << 16));              // td0 hi | td1 lo
  g1[3] = (int)((tensor_rows >> 16) | ((unsigned)DC << 16));   // td1 hi | tile_dim0=32
  g1[4] = (int)MT;                                             // tile_dim1=128 (tile_dim2=0)
  g1[5] = (int)DIM;                                            // tensor_dim0_stride lo32
  g1[6] = 0;                                                   // stride hi | dim1_stride lo
  g1[7] = 0;
  i32x4 zero4 = {0, 0, 0, 0};
  i32x8 zero8 = {0, 0, 0, 0, 0, 0, 0, 0};
  __builtin_amdgcn_tensor_load_to_lds(g0, g1, zero4, zero4, zero8, 0);
}

// fp32 WMMA GEMM (z @ embed^T) fused with running argmin of ||e||^2 - 2 z.e
// Global->LDS staging done by the Tensor Data Mover, double buffered.
__global__ void __launch_bounds__(256)
vq_dist(const float* __restrict__ z, const float* __restrict__ embed,
        const float* __restrict__ norms, float* __restrict__ cand_val,
        int* __restrict__ cand_idx) {
  __shared__ float lds_a[2][MT * ST];
  __shared__ float lds_b[2][NT * ST];
  const int tid  = threadIdx.x;
  const int lane = tid & 31;
  const int w    = tid >> 5;
  const int l15  = lane & 15;
  const int koff = (lane >> 4) << 1;           // K offset per half-wave (A & B layouts)
  const int rowbase  = blockIdx.x * MT;
  const int colbase0 = blockIdx.y * KSPLIT;

  const unsigned ldsA[2] = {(unsigned)(uintptr_t)&lds_a[0][0],
                            (unsigned)(uintptr_t)&lds_a[1][0]};
  const unsigned ldsB[2] = {(unsigned)(uintptr_t)&lds_b[0][0],
                            (unsigned)(uintptr_t)&lds_b[1][0]};
  const float* zA = z + (size_t)rowbase * DIM;

  float bestv[8];
  int   besti[8];
#pragma unroll
  for (int r = 0; r < 8; ++r) { bestv[r] = FLT_MAX; besti[r] = 0; }

  for (int cb = 0; cb < KSPLIT; cb += NT) {
    const int colbase = colbase0 + cb;
    const float* eB = embed + (size_t)colbase * DIM;
    v8f acc[8] = {};

    if (w == 0) {                               // TDM ignores EXEC; one DMA per wave issue
      tdm_load_tile(zA, ldsA[0], N_ROWS);
      tdm_load_tile(eB, ldsB[0], K_CODES);
    }

    for (int dci = 0; dci < NCHUNK; ++dci) {
      const int cur = dci & 1;
      if (w == 0) {
        if (dci + 1 < NCHUNK) {
          tdm_load_tile(zA + (dci + 1) * DC, ldsA[cur ^ 1], N_ROWS);
          tdm_load_tile(eB + (dci + 1) * DC, ldsB[cur ^ 1], K_CODES);
          __builtin_amdgcn_s_wait_tensorcnt((short)2);  // current chunk's 2 DMAs done
        } else {
          __builtin_amdgcn_s_wait_tensorcnt((short)0);
        }
      }
      __syncthreads();                          // all waves: chunk `cur` visible in LDS

      const float* ap = &lds_a[cur][(w * 16 + l15) * ST + koff];
      const float* bp = &lds_b[cur][l15 * ST + koff];
#pragma unroll
      for (int ks = 0; ks < DC; ks += 4) {
        v2f a = *(const v2f*)(ap + ks);
#pragma unroll
        for (int t = 0; t < 8; ++t) {
          v2f b = *(const v2f*)(bp + t * 16 * ST + ks);
          acc[t] = __builtin_amdgcn_wmma_f32_16x16x4_f32(
              false, a, false, b, (short)0, acc[t], false, false);
        }
      }
      __syncthreads();                          // done reading before buffer reuse
    }

    // fold this 128-column tile into running per-lane argmin
#pragma unroll
    for (int t = 0; t < 8; ++t) {
      const int col = colbase + t * 16 + l15;
      const float nb = norms[col];
#pragma unroll
      for (int r = 0; r < 8; ++r) {
        float d = nb - 2.0f * acc[t][r];
        if (d < bestv[r]) { bestv[r] = d; besti[r] = col; }  // strict <: first-min
      }
    }
  }

  // butterfly argmin across the 16 lanes sharing each row (wave32 halves)
#pragma unroll
  for (int r = 0; r < 8; ++r) {
#pragma unroll
    for (int m = 8; m >= 1; m >>= 1) {
      float ov = __shfl_xor(bestv[r], m, 32);
      int   oi = __shfl_xor(besti[r], m, 32);
      if (ov < bestv[r] || (ov == bestv[r] && oi < besti[r])) {
        bestv[r] = ov; besti[r] = oi;
      }
    }
    if (l15 == 0) {
      int row = rowbase + w * 16 + (lane >> 4) * 8 + r;  // C/D layout: lanes>=16 hold M+8
      cand_val[blockIdx.y * N_ROWS + row] = bestv[r];
      cand_idx[blockIdx.y * N_ROWS + row] = besti[r];
    }
  }
}

__global__ void vq_combine(const float* __restrict__ cand_val,
                           const int* __restrict__ cand_idx,
                           int* __restrict__ ws_idx, float* __restrict__ out_idxf) {
  int row = blockIdx.x * blockDim.x + threadIdx.x;
  if (row >= N_ROWS) return;
  float bv = cand_val[row];
  int   bi = cand_idx[row];
  for (int s = 1; s < NSPLIT; ++s) {
    float v = cand_val[s * N_ROWS + row];
    int   i = cand_idx[s * N_ROWS + row];
    if (v < bv || (v == bv && i < bi)) { bv = v; bi = i; }
  }
  ws_idx[row] = bi;
  out_idxf[row] = (float)bi;
}

// gather z_q = embed[idx], commitment-loss partials, histogram
__global__ void vq_gather(const float* __restrict__ z, const float* __restrict__ embed,
                          const int* __restrict__ ws_idx, float* __restrict__ out_zq,
                          unsigned* __restrict__ counts, float* __restrict__ loss) {
  const int w = threadIdx.x >> 5, lane = threadIdx.x & 31;
  const int row = blockIdx.x * 8 + w;
  const int idx = ws_idx[row];
  const float* e  = embed + (size_t)idx * DIM;
  const float* zr = z + (size_t)row * DIM;
  float* o = out_zq + (size_t)row * DIM;
  float s = 0.0f;
#pragma unroll
  for (int i = 0; i < 4; ++i) {
    int c = i * 128 + lane * 4;
    float4 ev = *(const float4*)(e + c);
    float4 zv = *(const float4*)(zr + c);
    *(float4*)(o + c) = ev;
    float dx = ev.x - zv.x, dy = ev.y - zv.y, dz = ev.z - zv.z, dw = ev.w - zv.w;
    s += dx * dx + dy * dy + dz * dz + dw * dw;
  }
#pragma unroll
  for (int m = 16; m >= 1; m >>= 1) s += __shfl_xor(s, m, 32);
  if (lane == 0) {
    atomicAdd(loss, s);
    atomicAdd(&counts[idx], 1u);
  }
}

__global__ void vq_final(const unsigned* __restrict__ counts,
                         const float* __restrict__ loss, float* __restrict__ out_scal) {
  __shared__ float red[256];
  const int tid = threadIdx.x;
  float s = 0.0f;
  for (int k = tid; k < K_CODES; k += 256) {
    float p = (float)counts[k] * (1.0f / (float)N_ROWS);
    s += p * logf(p + 1e-10f);
  }
  red[tid] = s;
  __syncthreads();
  for (int off = 128; off > 0; off >>= 1) {
    if (tid < off) red[tid] += red[tid + off];
    __syncthreads();
  }
  if (tid == 0) {
    out_scal[0] = 0.25f * loss[0] / ((float)N_ROWS * (float)DIM);  // vq_loss
    out_scal[1] = expf(-red[0]);                                   // perplexity
  }
}

extern "C" void kernel_launch(void* const* d_in, const int* in_sizes, int n_in,
                              void* d_out, int out_size, void* d_ws, size_t ws_size,
                              hipStream_t stream) {
  (void)in_sizes; (void)n_in; (void)out_size; (void)ws_size;
  const float* z     = (const float*)d_in[0];
  const float* embed = (const float*)d_in[1];
  float* out  = (float*)d_out;
  float* wsf  = (float*)d_ws;

  float*    norms    = wsf + WS_NORMS;
  unsigned* counts   = (unsigned*)(wsf + WS_COUNTS);
  float*    loss     = wsf + WS_LOSS;
  float*    cand_val = wsf + WS_CANDV;
  int*      cand_idx = (int*)(wsf + WS_CANDI);
  int*      ws_idx   = (int*)(wsf + WS_IDX);

  vq_init<<<(K_CODES + 255) / 256, 256, 0, stream>>>(counts, loss);
  vq_norms<<<K_CODES / 8, 256, 0, stream>>>(embed, norms);
  dim3 grid(N_ROWS / MT, NSPLIT);
  vq_dist<<<grid, 256, 0, stream>>>(z, embed, norms, cand_val, cand_idx);
  vq_combine<<<N_ROWS / 256, 256, 0, stream>>>(cand_val, cand_idx, ws_idx, out + OUT_IDX);
  vq_gather<<<N_ROWS / 8, 256, 0, stream>>>(z, embed, ws_idx, out, counts, loss);
  vq_final<<<1, 256, 0, stream>>>(counts, loss, out + OUT_SCAL);
}